// GAT_21268678050286
// MI455X (gfx1250) — compile-verified
//
#include <hip/hip_runtime.h>
#include <hip/hip_bf16.h>

typedef float v2f __attribute__((ext_vector_type(2)));
typedef float v8f __attribute__((ext_vector_type(8)));
typedef unsigned int u32x4 __attribute__((ext_vector_type(4)));
typedef int i32x4 __attribute__((ext_vector_type(4)));
typedef int i32x8 __attribute__((ext_vector_type(8)));

#define NEG_SLOPE 0.2f
#define ORD_NEG_INF 0x807FFFFF  // f2ord(-inf)

// Monotone int encoding of float so atomicMax(int) == float max.
__device__ __forceinline__ int f2ord(float f) {
    int i = __float_as_int(f);
    return (i < 0) ? (i ^ 0x7fffffff) : i;
}
__device__ __forceinline__ float ord2f(int i) {
    return __int_as_float((i < 0) ? (i ^ 0x7fffffff) : i);
}

__global__ void fill_i32_kernel(int* __restrict__ p, int v, int n) {
    int i = blockIdx.x * blockDim.x + threadIdx.x;
    if (i < n) p[i] = v;
}

// C[M,Ncols] = A[M,K] * B[K,Ncols] in fp32 using V_WMMA_F32_16X16X4_F32.
// block = 128 threads (4 waves). Wave w computes the 16x16 tile at
// (blockIdx.x*16, (blockIdx.y*4+w)*16). Requires M%16==0, K%4==0, Ncols%16==0,
// and exact grid coverage (no partial tiles) so EXEC stays all-1s for WMMA.
//
// The 16xK fp32 A-tile is DMA'd into LDS once per block by the Tensor Data
// Mover (tensor_load_to_lds). The TDM pad feature inserts 1 DWORD after every
// 128 DWORDs, giving a 129-DWORD LDS row stride => the 16-lane column reads
// map to banks (row + k) % 64: conflict-free.
//
// WMMA fragment layouts (ISA 7.12.2, fp32 K=4):
//   A 16x4 : lane<16 holds K={0,1}, lane>=16 holds K={2,3} of row (lane&15).
//   B 4x16 : lane<16 holds K={0,1}, lane>=16 holds K={2,3} of col (lane&15).
//   D      : VGPR r = row (r + 8*(lane>>4)), col (lane&15).
#define LDS_ROW 129
__global__ void gemm_wmma_f32(const float* __restrict__ A,
                              const float* __restrict__ B,
                              float* __restrict__ C,
                              int M, int K, int Ncols) {
    __shared__ float As[16 * LDS_ROW + 8];  // padded 16 x K tile (K<=128)
    const int lane = threadIdx.x & 31;
    const int wave = threadIdx.x >> 5;
    const int m0 = blockIdx.x << 4;
    const int n0 = ((blockIdx.y << 2) + wave) << 4;

    if (wave == 0) {
        // ---- Tensor DMA descriptor (CDNA5 ISA) ----
        const unsigned lds = (unsigned)(size_t)(void*)As;  // LDS byte offset
        const unsigned long long ga = (unsigned long long)(A + (size_t)m0 * K);
        u32x4 g0;
        g0[0] = 1u;                                        // count=1 (valid D#)
        g0[1] = lds;                                       // lds_addr
        g0[2] = (unsigned)(ga & 0xffffffffu);              // global_addr[31:0]
        g0[3] = (unsigned)((ga >> 32) & 0x01ffffffu)       // global_addr[56:32]
              | (2u << 30);                                // type=2 ("image")
        i32x8 g1;
        g1[0] = (int)((2u << 16)     // data_size=2 -> 4-byte elements
                    | (1u << 20)     // pad_enable
                    | (6u << 22));   // pad_interval=6 -> every 128 DWORDs
                                     // pad_amount=0  -> 1 DWORD
        g1[1] = (int)(((unsigned)K & 0xffffu) << 16);      // tensor_dim0 lo16
        g1[2] = (int)((((unsigned)K >> 16) & 0xffffu)      // tensor_dim0 hi16
                    | (16u << 16));                        // tensor_dim1 = 16
        g1[3] = (int)(((unsigned)K & 0xffffu) << 16);      // tile_dim0 = K
        g1[4] = 16;                                        // tile_dim1 = 16
        g1[5] = K;                                         // tensor_dim0_stride lo32
        g1[6] = 0;
        g1[7] = 0;
        i32x4 gz4 = {0, 0, 0, 0};                          // groups 2/3 off (2-D)
        i32x8 gz8 = {0, 0, 0, 0, 0, 0, 0, 0};
        __builtin_amdgcn_tensor_load_to_lds(g0, g1, gz4, gz4, gz8, 0);
        __builtin_amdgcn_s_wait_tensorcnt(0);              // TENSORcnt is per-wave
    }
    __syncthreads();                                       // publish LDS tile

    const int l15 = lane & 15;
    const int hi  = lane >> 4;
    const float* __restrict__ Ar = As + l15 * LDS_ROW;     // LDS, conflict-free
    const float* __restrict__ Bc = B + n0 + l15;
    v8f acc = {};
    for (int kb = 0; kb < K; kb += 4) {
        const int ka = kb + hi * 2;
        v2f a, b;
        a[0] = Ar[ka];
        a[1] = Ar[ka + 1];
        b[0] = Bc[(size_t)ka * Ncols];
        b[1] = Bc[(size_t)(ka + 1) * Ncols];
        acc = __builtin_amdgcn_wmma_f32_16x16x4_f32(false, a, false, b,
                                                    (short)0, acc, false, false);
    }
#pragma unroll
    for (int r = 0; r < 8; ++r)
        C[(size_t)(m0 + r + 8 * hi) * Ncols + n0 + l15] = acc[r];
}

// a_src[n,h] = dot(h1[n, h*16 .. h*16+15], att_src[h,:]); likewise a_dst.
__global__ void att1_kernel(const float* __restrict__ h1,
                            const float* __restrict__ att_src,
                            const float* __restrict__ att_dst,
                            float* __restrict__ a_s, float* __restrict__ a_d,
                            int NH) {
    int t = blockIdx.x * blockDim.x + threadIdx.x;
    if (t >= NH) return;
    int n = t >> 3, h = t & 7;
    const float* hp = h1 + (size_t)n * 128 + h * 16;
    const float* as = att_src + h * 16;
    const float* ad = att_dst + h * 16;
    float s = 0.f, d = 0.f;
#pragma unroll
    for (int i = 0; i < 16; ++i) {
        float v = hp[i];
        s += v * as[i];
        d += v * ad[i];
    }
    a_s[t] = s;
    a_d[t] = d;
}

// a2[n] = dot(h2[n,:64], att[0,:64])
__global__ void att2_kernel(const float* __restrict__ h2,
                            const float* __restrict__ att_src,
                            const float* __restrict__ att_dst,
                            float* __restrict__ a_s, float* __restrict__ a_d,
                            int N) {
    int n = blockIdx.x * blockDim.x + threadIdx.x;
    if (n >= N) return;
    const float* hp = h2 + (size_t)n * 64;
    float s = 0.f, d = 0.f;
#pragma unroll 8
    for (int i = 0; i < 64; ++i) {
        float v = hp[i];
        s += v * att_src[i];
        d += v * att_dst[i];
    }
    a_s[n] = s;
    a_d[n] = d;
}

// Pass 1 (layer 1): e = leaky_relu(a_src[src]+a_dst[dst]); store e; segment max.
__global__ void edge1_max_kernel(const int* __restrict__ src, const int* __restrict__ dst,
                                 const float* __restrict__ a_s, const float* __restrict__ a_d,
                                 float* __restrict__ e1, int* __restrict__ m1, int E) {
    int e = blockIdx.x * blockDim.x + threadIdx.x;
    if (e >= E) return;
    int s = src[e], d = dst[e];
#pragma unroll
    for (int h = 0; h < 8; ++h) {
        float v = a_s[s * 8 + h] + a_d[d * 8 + h];
        v = (v > 0.f) ? v : NEG_SLOPE * v;
        e1[(size_t)e * 8 + h] = v;
        atomicMax(m1 + d * 8 + h, f2ord(v));
    }
}

// Pass 2 (layer 1): ee = exp(e - m[dst]); store ee; segment sum.
__global__ void edge1_exp_kernel(const int* __restrict__ dst,
                                 float* __restrict__ e1, const int* __restrict__ m1,
                                 float* __restrict__ dn1, int E) {
    int e = blockIdx.x * blockDim.x + threadIdx.x;
    if (e >= E) return;
    int d = dst[e];
#pragma unroll
    for (int h = 0; h < 8; ++h) {
        float v  = e1[(size_t)e * 8 + h];
        float mm = ord2f(m1[d * 8 + h]);
        float ee = __expf(v - mm);
        e1[(size_t)e * 8 + h] = ee;
        atomicAdd(dn1 + d * 8 + h, ee);
    }
}

// Pass 3 (layer 1): one wave per edge; lane moves float4 of the 128-float row.
// head = (lane*4)/16 = lane>>2. Destination rows (51 MB) are L2-resident.
__global__ void edge1_msg_kernel(const int* __restrict__ src, const int* __restrict__ dst,
                                 const float* __restrict__ h1, const float* __restrict__ e1,
                                 const float* __restrict__ dn1, float* __restrict__ acc, int E) {
    int gid  = blockIdx.x * blockDim.x + threadIdx.x;
    int edge = gid >> 5;
    int lane = threadIdx.x & 31;
    if (edge >= E) return;
    int s = src[edge], d = dst[edge];
    int head = lane >> 2;
    float alpha = e1[(size_t)edge * 8 + head] / (dn1[d * 8 + head] + 1e-16f);
    const float4 hv = *(const float4*)(h1 + (size_t)s * 128 + lane * 4);
    float* o = acc + (size_t)d * 128 + lane * 4;
    atomicAdd(o + 0, hv.x * alpha);
    atomicAdd(o + 1, hv.y * alpha);
    atomicAdd(o + 2, hv.z * alpha);
    atomicAdd(o + 3, hv.w * alpha);
}

// In-place: x2 = elu(acc + b1[c])
__global__ void elu_bias_kernel(float* __restrict__ acc, const float* __restrict__ b1, int n) {
    int i = blockIdx.x * blockDim.x + threadIdx.x;
    if (i >= n) return;
    float v = acc[i] + b1[i & 127];
    acc[i] = (v > 0.f) ? v : expm1f(v);
}

// Layer-2 edge passes (H=1)
__global__ void edge2_max_kernel(const int* __restrict__ src, const int* __restrict__ dst,
                                 const float* __restrict__ a_s, const float* __restrict__ a_d,
                                 float* __restrict__ e2, int* __restrict__ m2, int E) {
    int e = blockIdx.x * blockDim.x + threadIdx.x;
    if (e >= E) return;
    float v = a_s[src[e]] + a_d[dst[e]];
    v = (v > 0.f) ? v : NEG_SLOPE * v;
    e2[e] = v;
    atomicMax(m2 + dst[e], f2ord(v));
}

__global__ void edge2_exp_kernel(const int* __restrict__ dst,
                                 float* __restrict__ e2, const int* __restrict__ m2,
                                 float* __restrict__ dn2, int E) {
    int e = blockIdx.x * blockDim.x + threadIdx.x;
    if (e >= E) return;
    int d = dst[e];
    float ee = __expf(e2[e] - ord2f(m2[d]));
    e2[e] = ee;
    atomicAdd(dn2 + d, ee);
}

// Half-wave (16 lanes x float4 = 64 floats) per edge; wave handles 2 edges.
__global__ void edge2_msg_kernel(const int* __restrict__ src, const int* __restrict__ dst,
                                 const float* __restrict__ h2, const float* __restrict__ e2,
                                 const float* __restrict__ dn2, float* __restrict__ out, int E) {
    int gid  = blockIdx.x * blockDim.x + threadIdx.x;
    int pair = gid >> 5;
    int lane = threadIdx.x & 31;
    int edge = pair * 2 + (lane >> 4);
    if (edge >= E) return;
    int s = src[edge], d = dst[edge];
    int l15 = lane & 15;
    float alpha = e2[edge] / (dn2[d] + 1e-16f);
    const float4 hv = *(const float4*)(h2 + (size_t)s * 64 + l15 * 4);
    float* o = out + (size_t)d * 64 + l15 * 4;
    atomicAdd(o + 0, hv.x * alpha);
    atomicAdd(o + 1, hv.y * alpha);
    atomicAdd(o + 2, hv.z * alpha);
    atomicAdd(o + 3, hv.w * alpha);
}

__global__ void bias2_kernel(float* __restrict__ out, const float* __restrict__ b2, int n) {
    int i = blockIdx.x * blockDim.x + threadIdx.x;
    if (i >= n) return;
    out[i] += b2[i & 63];
}

static inline int cdiv(long long a, long long b) { return (int)((a + b - 1) / b); }

extern "C" void kernel_launch(void* const* d_in, const int* in_sizes, int n_in,
                              void* d_out, int out_size, void* d_ws, size_t ws_size,
                              hipStream_t stream) {
    const float* x        = (const float*)d_in[0];
    const int*   eidx     = (const int*)  d_in[1];
    const float* W1       = (const float*)d_in[2];
    const float* att_src1 = (const float*)d_in[3];
    const float* att_dst1 = (const float*)d_in[4];
    const float* b1       = (const float*)d_in[5];
    const float* W2       = (const float*)d_in[6];
    const float* att_src2 = (const float*)d_in[7];
    const float* att_dst2 = (const float*)d_in[8];
    const float* b2       = (const float*)d_in[9];

    const int N = in_sizes[0] / 128;  // 100000
    const int E = in_sizes[1] / 2;    // 1600000
    const int* src = eidx;
    const int* dst = eidx + E;

    // ---- workspace carve-out (with reuse) ----
    char* ws = (char*)d_ws;
    size_t off = 0;
    auto take = [&](size_t bytes) {
        void* p = ws + off;
        off = (off + bytes + 255) & ~(size_t)255;
        return p;
    };
    float* h1   = (float*)take((size_t)N * 128 * 4);  // layer1 pre-act; reused as h2 (N*64)
    float* acc1 = (float*)take((size_t)N * 128 * 4);  // layer1 aggregate; becomes x2 (in-place ELU)
    float* e1   = (float*)take((size_t)E * 8 * 4);    // per-edge e/ee (layer1); reused as e2 (E)
    float* a1s  = (float*)take((size_t)N * 8 * 4);
    float* a1d  = (float*)take((size_t)N * 8 * 4);
    int*   m1   = (int*)  take((size_t)N * 8 * 4);
    float* dn1  = (float*)take((size_t)N * 8 * 4);
    float* a2s  = (float*)take((size_t)N * 4);
    float* a2d  = (float*)take((size_t)N * 4);
    int*   m2   = (int*)  take((size_t)N * 4);
    float* dn2  = (float*)take((size_t)N * 4);
    float* h2   = h1;
    float* e2   = e1;
    float* outp = (float*)d_out;
    (void)ws_size; (void)n_in;

    // ---- init (all capturable) ----
    (void)hipMemsetAsync(acc1, 0, (size_t)N * 128 * 4, stream);
    (void)hipMemsetAsync(dn1,  0, (size_t)N * 8 * 4,  stream);
    (void)hipMemsetAsync(dn2,  0, (size_t)N * 4,      stream);
    (void)hipMemsetAsync(d_out, 0, (size_t)out_size * 4, stream);
    fill_i32_kernel<<<cdiv((long long)N * 8, 256), 256, 0, stream>>>(m1, ORD_NEG_INF, N * 8);
    fill_i32_kernel<<<cdiv(N, 256), 256, 0, stream>>>(m2, ORD_NEG_INF, N);

    // ---- layer 1 ----
    // h1 = x @ W1   (WMMA fp32, A tile staged via TDM)
    gemm_wmma_f32<<<dim3(N / 16, 2), 128, 0, stream>>>(x, W1, h1, N, 128, 128);
    att1_kernel<<<cdiv((long long)N * 8, 256), 256, 0, stream>>>(h1, att_src1, att_dst1, a1s, a1d, N * 8);
    edge1_max_kernel<<<cdiv(E, 256), 256, 0, stream>>>(src, dst, a1s, a1d, e1, m1, E);
    edge1_exp_kernel<<<cdiv(E, 256), 256, 0, stream>>>(dst, e1, m1, dn1, E);
    edge1_msg_kernel<<<cdiv((long long)E * 32, 256), 256, 0, stream>>>(src, dst, h1, e1, dn1, acc1, E);
    elu_bias_kernel<<<cdiv((long long)N * 128, 256), 256, 0, stream>>>(acc1, b1, N * 128);

    // ---- layer 2 ----
    // h2 = x2 @ W2  (WMMA fp32); x2 == acc1 (post-ELU)
    gemm_wmma_f32<<<dim3(N / 16, 1), 128, 0, stream>>>(acc1, W2, h2, N, 128, 64);
    att2_kernel<<<cdiv(N, 256), 256, 0, stream>>>(h2, att_src2, att_dst2, a2s, a2d, N);
    edge2_max_kernel<<<cdiv(E, 256), 256, 0, stream>>>(src, dst, a2s, a2d, e2, m2, E);
    edge2_exp_kernel<<<cdiv(E, 256), 256, 0, stream>>>(dst, e2, m2, dn2, E);
    edge2_msg_kernel<<<cdiv((long long)(E / 2) * 32, 256), 256, 0, stream>>>(src, dst, h2, e2, dn2, outp, E);
    bias2_kernel<<<cdiv((long long)N * 64, 256), 256, 0, stream>>>(outp, b2, N * 64);
}